// _ECELoss_1657857376954
// MI455X (gfx1250) — compile-verified
//
#include <hip/hip_runtime.h>
#include <stdint.h>

// Problem constants (match reference)
constexpr int B_ROWS   = 2048;
constexpr int S_POS    = 3;
constexpr int V_VOCAB  = 32000;
constexpr int ROWS     = B_ROWS * S_POS;        // 6144 softmax rows
constexpr int THREADS  = 256;                   // 8 wave32 per WG
constexpr int STAGE_FL = THREADS * 4;           // 1024 floats (4 KB) per async stage
constexpr int NSTAGES  = 31;                    // 31*1024 = 31744 floats via async pipe
constexpr int TAIL     = NSTAGES * STAGE_FL;    // 31744; tail = 256 floats (1/thread)
constexpr int DEPTH    = 4;                     // async pipeline depth (16 KB LDS)
constexpr int NBINS    = 15;

__device__ __forceinline__ unsigned lds_off(const void* p) {
    // generic (flat) pointer into the LDS aperture: low 32 bits are the
    // wave-relative LDS byte offset (CDNA5 flat->LDS mapping truncates addr[31:0])
    return (unsigned)(unsigned long long)p;
}

__device__ __forceinline__ void online_update(float x, int col, float& m, float& d, int& idx) {
    float mn = fmaxf(m, x);
    d = d * __expf(m - mn) + __expf(x - mn);
    idx = (x > m) ? col : idx;
    m = mn;
}

__global__ __launch_bounds__(THREADS) void row_softmax_stats(
    const float* __restrict__ logits,
    float* __restrict__ conf_out,
    int*   __restrict__ pred_out)
{
    __shared__ __align__(16) float stage[DEPTH * STAGE_FL];   // 16 KB quad buffer
    __shared__ float red_m[THREADS / 32];
    __shared__ float red_d[THREADS / 32];
    __shared__ int   red_i[THREADS / 32];

    const int row = blockIdx.x;
    const int tid = threadIdx.x;
    const float* rowp = logits + (size_t)row * V_VOCAB;       // uniform -> SGPR pair

    const unsigned go0   = (unsigned)tid * 16u;               // lane byte offset in row
    const unsigned lbase = lds_off(&stage[0]) + (unsigned)tid * 16u;

    // ---- prologue: fill the 4-deep async pipeline (stages 0..3) ----
    asm volatile("global_load_async_to_lds_b128 %0, %1, %2 offset:0"
                 :: "v"(lbase + 0u * 4096u), "v"(go0), "s"(rowp) : "memory");
    asm volatile("global_load_async_to_lds_b128 %0, %1, %2 offset:4096"
                 :: "v"(lbase + 1u * 4096u), "v"(go0), "s"(rowp) : "memory");
    asm volatile("global_load_async_to_lds_b128 %0, %1, %2 offset:8192"
                 :: "v"(lbase + 2u * 4096u), "v"(go0), "s"(rowp) : "memory");
    asm volatile("global_load_async_to_lds_b128 %0, %1, %2 offset:12288"
                 :: "v"(lbase + 3u * 4096u), "v"(go0), "s"(rowp) : "memory");

    float m = -INFINITY, d = 0.0f;
    int   idx = 0;

    for (int k = 0; k < NSTAGES; ++k) {
        // retire the oldest stage (async loads complete in order)
        const int rem = (NSTAGES - 1) - k;   // stages still legitimately in flight
        if (rem >= 3)      asm volatile("s_wait_asynccnt 0x3" ::: "memory");
        else if (rem == 2) asm volatile("s_wait_asynccnt 0x2" ::: "memory");
        else if (rem == 1) asm volatile("s_wait_asynccnt 0x1" ::: "memory");
        else               asm volatile("s_wait_asynccnt 0x0" ::: "memory");

        // each lane consumes exactly the 16 B it copied -> no workgroup barrier
        const int buf = k & (DEPTH - 1);
        float4 x4 = *reinterpret_cast<const float4*>(&stage[buf * STAGE_FL + tid * 4]);

        // buffer is free once the ds_load has drained; then refill it
        asm volatile("s_wait_dscnt 0x0" ::: "memory");
        if (k + DEPTH < NSTAGES) {
            unsigned go = go0 + (unsigned)(k + DEPTH) * 4096u;
            asm volatile("global_load_async_to_lds_b128 %0, %1, %2 offset:0"
                         :: "v"(lbase + (unsigned)buf * 4096u), "v"(go), "s"(rowp)
                         : "memory");
        }

        const int colbase = k * STAGE_FL + tid * 4;
        online_update(x4.x, colbase + 0, m, d, idx);
        online_update(x4.y, colbase + 1, m, d, idx);
        online_update(x4.z, colbase + 2, m, d, idx);
        online_update(x4.w, colbase + 3, m, d, idx);
    }

    // tail: remaining 256 floats, one per thread
    {
        float x = rowp[TAIL + tid];
        online_update(x, TAIL + tid, m, d, idx);
    }

    // wave32 shuffle reduction of (m, d, idx) -- fixed order, deterministic
    #pragma unroll
    for (int off = 16; off >= 1; off >>= 1) {
        float om = __shfl_down(m, off, 32);
        float od = __shfl_down(d, off, 32);
        int   oi = __shfl_down(idx, off, 32);
        float mn = fmaxf(m, om);
        d   = d * __expf(m - mn) + od * __expf(om - mn);
        idx = (om > m) ? oi : idx;
        m   = mn;
    }

    const int wave = tid >> 5, lane = tid & 31;
    if (lane == 0) { red_m[wave] = m; red_d[wave] = d; red_i[wave] = idx; }
    __syncthreads();

    if (tid == 0) {
        float M = red_m[0], D = red_d[0];
        int   I = red_i[0];
        #pragma unroll
        for (int w = 1; w < THREADS / 32; ++w) {
            float om = red_m[w], od = red_d[w];
            int   oi = red_i[w];
            float mn = fmaxf(M, om);
            D = D * __expf(M - mn) + od * __expf(om - mn);
            I = (om > M) ? oi : I;
            M = mn;
        }
        conf_out[row] = 1.0f / D;   // max softmax prob = exp(M-M)/Z
        pred_out[row] = I;
    }
}

// Deterministic bin reduction: private per-thread accumulators -> LDS partial
// matrix -> fixed-order sequential folds. No float atomics anywhere.
__global__ __launch_bounds__(256) void ece_reduce(
    const float*     __restrict__ conf3,
    const int*       __restrict__ pred3,
    const long long* __restrict__ targets,   // int64 (B,4)
    float*           __restrict__ out)
{
    __shared__ float s_cnt [NBINS][256];   // 15 KB
    __shared__ float s_conf[NBINS][256];   // 15 KB
    __shared__ float s_acc [NBINS][256];   // 15 KB
    __shared__ float s_term[NBINS];

    const int tid = threadIdx.x;

    float p_cnt[NBINS], p_conf[NBINS], p_acc[NBINS];
    #pragma unroll
    for (int i = 0; i < NBINS; ++i) { p_cnt[i] = 0.f; p_conf[i] = 0.f; p_acc[i] = 0.f; }

    for (int b = tid; b < B_ROWS; b += 256) {
        float c = conf3[b * 3 + 0] * conf3[b * 3 + 1] * conf3[b * 3 + 2];
        float acc = 0.f;
        #pragma unroll
        for (int s = 0; s < 3; ++s)
            acc += (pred3[b * 3 + s] == (int)targets[b * 4 + 1 + s]) ? 1.f : 0.f;
        acc *= (1.0f / 3.0f);

        #pragma unroll
        for (int i = 0; i < NBINS; ++i) {
            float lo = (float)i       / (float)NBINS;
            float hi = (float)(i + 1) / (float)NBINS;
            bool in  = (c > lo) && (c <= hi);
            p_cnt[i]  += in ? 1.0f : 0.0f;
            p_conf[i] += in ? c    : 0.0f;
            p_acc[i]  += in ? acc  : 0.0f;
        }
    }

    #pragma unroll
    for (int i = 0; i < NBINS; ++i) {
        s_cnt [i][tid] = p_cnt[i];
        s_conf[i][tid] = p_conf[i];
        s_acc [i][tid] = p_acc[i];
    }
    __syncthreads();

    if (tid < NBINS) {
        float cnt = 0.f, sc = 0.f, sa = 0.f;
        for (int j = 0; j < 256; ++j) {        // fixed order -> deterministic
            cnt += s_cnt [tid][j];
            sc  += s_conf[tid][j];
            sa  += s_acc [tid][j];
        }
        float prop = cnt / (float)B_ROWS;
        float safe = fmaxf(cnt, 1.0f);
        s_term[tid] = (cnt > 0.f) ? fabsf(sc / safe - sa / safe) * prop : 0.f;
    }
    __syncthreads();

    if (tid == 0) {
        float e = 0.f;
        #pragma unroll
        for (int i = 0; i < NBINS; ++i) e += s_term[i];
        out[0] = e;
    }
}

extern "C" void kernel_launch(void* const* d_in, const int* in_sizes, int n_in,
                              void* d_out, int out_size, void* d_ws, size_t ws_size,
                              hipStream_t stream)
{
    (void)in_sizes; (void)n_in; (void)out_size; (void)ws_size;
    const float*     logits  = (const float*)d_in[0];
    const long long* targets = (const long long*)d_in[1];

    float* conf3 = (float*)d_ws;                               // 6144 floats
    int*   pred3 = (int*)((char*)d_ws + ROWS * sizeof(float)); // 6144 ints

    row_softmax_stats<<<ROWS, THREADS, 0, stream>>>(logits, conf3, pred3);
    ece_reduce<<<1, 256, 0, stream>>>(conf3, pred3, targets, (float*)d_out);
}